// GlobalUpdate_91096256348932
// MI455X (gfx1250) — compile-verified
//
#include <hip/hip_runtime.h>
#include <cfloat>

// ---------------------------------------------------------------------------
// GNN global update for MI455X (gfx1250, wave32).
// Segment reductions are HBM-bound (~350MB -> ~15us @ 23.3TB/s); the MLP is
// tiny and runs on V_WMMA_F32_16X16X4_F32 (full f32 precision, no downcast).
// ---------------------------------------------------------------------------

#define G_NUM 128
#define DFEAT 64
#define GD (G_NUM * DFEAT)      // 8192
#define N_IN 576
#define N_HID 256
#define N_OUT 64

typedef __attribute__((ext_vector_type(2))) float v2f;
typedef __attribute__((ext_vector_type(8))) float v8f;

// Scratch layout (float offsets into d_ws):
//   0      emin[8192]   init +FLT_MAX
//   8192   esum[8192]   init 0
//   16384  emax[8192]   init -FLT_MAX
//   24576  ecnt[128]    init 0
//   24704  vmin[8192]
//   32896  vsum[8192]
//   41088  vmax[8192]
//   49280  vcnt[128]
//   49408  X[128*576]
//   123136 H[128*256]
#define OFF_EMIN 0
#define OFF_ESUM 8192
#define OFF_EMAX 16384
#define OFF_ECNT 24576
#define OFF_VMIN 24704
#define OFF_VSUM 32896
#define OFF_VMAX 41088
#define OFF_VCNT 49280
#define OFF_X    49408
#define OFF_H    123136
#define STATS_FLOATS 49408

// Float min/max atomics via the signed/unsigned integer ordering trick
// (exact for non-NaN data; lowers to ds_min_i32 / global_atomic_min_i32 etc.)
__device__ __forceinline__ void atomicMinFloat(float* a, float v) {
  int iv = __float_as_int(v);
  if (iv >= 0) atomicMin((int*)a, iv);
  else         atomicMax((unsigned int*)a, __float_as_uint(v));
}
__device__ __forceinline__ void atomicMaxFloat(float* a, float v) {
  int iv = __float_as_int(v);
  if (iv >= 0) atomicMax((int*)a, iv);
  else         atomicMin((unsigned int*)a, __float_as_uint(v));
}

// ---------------------------------------------------------------------------
// Init stats region of scratch (d_ws is poisoned by the harness every run).
// 49408 = 193 * 256 exactly.
__global__ void init_stats_kernel(float* __restrict__ ws) {
  int idx = blockIdx.x * blockDim.x + threadIdx.x;
  if (idx >= STATS_FLOATS) return;
  int m = idx % (OFF_VMIN);            // same layout in e-half and v-half
  float v;
  if (m < OFF_ESUM)      v = FLT_MAX;  // min identity
  else if (m < OFF_EMAX) v = 0.0f;     // sum identity
  else if (m < OFF_ECNT) v = -FLT_MAX; // max identity
  else                   v = 0.0f;     // count
  ws[idx] = v;
}

// ---------------------------------------------------------------------------
// Streaming segment min/sum/max/count. One wave per row: lane t handles
// features 2t,2t+1 (float2 -> 256B coalesced row load, LDS banks 2t..2t+1 ->
// conflict-free). All 128 graphs privatized in 96KB of LDS (CDNA5: 320KB/WGP),
// then one global-atomic merge per block.
__global__ void seg_reduce_kernel(const float* __restrict__ x,   // [M, 64]
                                  const int* __restrict__ batch, // [N]
                                  const int* __restrict__ src,   // edge src ids or nullptr
                                  int M,
                                  float* __restrict__ gmin,
                                  float* __restrict__ gsum,
                                  float* __restrict__ gmax,
                                  float* __restrict__ gcnt) {
  __shared__ float smin[GD];
  __shared__ float ssum[GD];
  __shared__ float smax[GD];
  __shared__ float scnt[G_NUM];

  const int tid = threadIdx.x;
  for (int i = tid; i < GD; i += blockDim.x) {
    smin[i] = FLT_MAX; ssum[i] = 0.0f; smax[i] = -FLT_MAX;
  }
  for (int i = tid; i < G_NUM; i += blockDim.x) scnt[i] = 0.0f;
  __syncthreads();

  const int lane   = tid & 31;
  const int wave   = tid >> 5;
  const int wpb    = blockDim.x >> 5;
  const int gwave  = blockIdx.x * wpb + wave;
  const int nwaves = gridDim.x * wpb;
  const int d0     = lane * 2;

  for (int row = gwave; row < M; row += nwaves) {
    const int s = src ? batch[src[row]] : batch[row];
    const float2 v = *(const float2*)(x + (size_t)row * DFEAT + d0);
    const int base = s * DFEAT + d0;
    atomicMinFloat(&smin[base],     v.x);
    atomicMinFloat(&smin[base + 1], v.y);
    atomicAdd(&ssum[base],     v.x);
    atomicAdd(&ssum[base + 1], v.y);
    atomicMaxFloat(&smax[base],     v.x);
    atomicMaxFloat(&smax[base + 1], v.y);
    if (lane == 0) atomicAdd(&scnt[s], 1.0f);
  }
  __syncthreads();

  for (int i = tid; i < GD; i += blockDim.x) {
    const float mn = smin[i];
    if (mn < FLT_MAX)  atomicMinFloat(&gmin[i], mn);
    const float mx = smax[i];
    if (mx > -FLT_MAX) atomicMaxFloat(&gmax[i], mx);
    const float sm = ssum[i];
    if (sm != 0.0f)    atomicAdd(&gsum[i], sm);
  }
  for (int i = tid; i < G_NUM; i += blockDim.x) {
    const float c = scnt[i];
    if (c != 0.0f) atomicAdd(&gcnt[i], c);
  }
}

// ---------------------------------------------------------------------------
// X[g, :] = [ g_attr | e_min e_mean e_sum e_max | v_min v_mean v_sum v_max ]
__global__ void build_x_kernel(const float* __restrict__ gattr,
                               const float* __restrict__ ws,
                               float* __restrict__ X) {
  const int idx = blockIdx.x * blockDim.x + threadIdx.x;  // 0..8191
  const int g = idx >> 6;
  const int d = idx & 63;
  const int gd = g * DFEAT + d;
  const float ec = fmaxf(ws[OFF_ECNT + g], 1.0f);
  const float vc = fmaxf(ws[OFF_VCNT + g], 1.0f);
  float* xr = X + g * N_IN;
  const float esum = ws[OFF_ESUM + gd];
  const float vsum = ws[OFF_VSUM + gd];
  xr[d]         = gattr[gd];
  xr[64 + d]    = ws[OFF_EMIN + gd];
  xr[128 + d]   = esum / ec;
  xr[192 + d]   = esum;
  xr[256 + d]   = ws[OFF_EMAX + gd];
  xr[320 + d]   = ws[OFF_VMIN + gd];
  xr[384 + d]   = vsum / vc;
  xr[448 + d]   = vsum;
  xr[512 + d]   = ws[OFF_VMAX + gd];
}

// ---------------------------------------------------------------------------
// H = relu(X @ W1 + b1).  One wave per 16x16 tile; 8x16 tiles; K=576 by 4.
// ISA 7.12.2 layouts: A lanes<16 hold (k,k+1), lanes>=16 hold (k+2,k+3);
// B VGPR0 holds rows k,k+2, VGPR1 rows k+1,k+3; C VGPR r -> rows r / r+8.
__global__ void gemm1_relu_kernel(const float* __restrict__ X,
                                  const float* __restrict__ W1,
                                  const float* __restrict__ b1,
                                  float* __restrict__ H) {
  const int gw   = (blockIdx.x * blockDim.x + threadIdx.x) >> 5; // 0..127
  const int lane = threadIdx.x & 31;
  const int mt = gw >> 4, nt = gw & 15;
  const int m0 = mt * 16, n0 = nt * 16;
  const int l  = lane & 15;
  const int hi = lane >> 4;

  v8f c = {0.f, 0.f, 0.f, 0.f, 0.f, 0.f, 0.f, 0.f};
  const float* xrow = X + (m0 + l) * N_IN;
  const float* wcol = W1 + n0 + l;
#pragma unroll 4
  for (int k = 0; k < N_IN; k += 4) {
    const int ka = k + hi * 2;
    v2f a, b;
    a.x = xrow[ka];
    a.y = xrow[ka + 1];
    b.x = wcol[(size_t)ka * N_HID];
    b.y = wcol[(size_t)(ka + 1) * N_HID];
    c = __builtin_amdgcn_wmma_f32_16x16x4_f32(false, a, false, b,
                                              (short)0, c, false, false);
  }
#pragma unroll
  for (int r = 0; r < 8; ++r) {
    const int m = m0 + r + hi * 8;
    const int n = n0 + l;
    H[m * N_HID + n] = fmaxf(c[r] + b1[n], 0.0f);
  }
}

// ---------------------------------------------------------------------------
// OUT = H @ W2 + b2.  8x4 tiles = 32 waves; K=256 by 4.
__global__ void gemm2_kernel(const float* __restrict__ H,
                             const float* __restrict__ W2,
                             const float* __restrict__ b2,
                             float* __restrict__ out) {
  const int gw   = (blockIdx.x * blockDim.x + threadIdx.x) >> 5; // 0..31
  const int lane = threadIdx.x & 31;
  const int mt = gw >> 2, nt = gw & 3;
  const int m0 = mt * 16, n0 = nt * 16;
  const int l  = lane & 15;
  const int hi = lane >> 4;

  v8f c = {0.f, 0.f, 0.f, 0.f, 0.f, 0.f, 0.f, 0.f};
  const float* hrow = H + (m0 + l) * N_HID;
  const float* wcol = W2 + n0 + l;
#pragma unroll 4
  for (int k = 0; k < N_HID; k += 4) {
    const int ka = k + hi * 2;
    v2f a, b;
    a.x = hrow[ka];
    a.y = hrow[ka + 1];
    b.x = wcol[(size_t)ka * N_OUT];
    b.y = wcol[(size_t)(ka + 1) * N_OUT];
    c = __builtin_amdgcn_wmma_f32_16x16x4_f32(false, a, false, b,
                                              (short)0, c, false, false);
  }
#pragma unroll
  for (int r = 0; r < 8; ++r) {
    const int m = m0 + r + hi * 8;
    const int n = n0 + l;
    out[m * N_OUT + n] = c[r] + b2[n];
  }
}

// ---------------------------------------------------------------------------
extern "C" void kernel_launch(void* const* d_in, const int* in_sizes, int n_in,
                              void* d_out, int out_size, void* d_ws, size_t ws_size,
                              hipStream_t stream) {
  const float* v_attr = (const float*)d_in[0];   // [N, 64]
  const int*   edgeij = (const int*)d_in[1];     // [2, E] row-major
  const float* e_attr = (const float*)d_in[2];   // [E, 64]
  const float* gattr  = (const float*)d_in[3];   // [128, 64]
  const int*   batch  = (const int*)d_in[4];     // [N]
  const float* W1     = (const float*)d_in[5];   // [576, 256]
  const float* b1     = (const float*)d_in[6];   // [256]
  const float* W2     = (const float*)d_in[7];   // [256, 64]
  const float* b2     = (const float*)d_in[8];   // [64]
  float* out = (float*)d_out;

  const int N = in_sizes[0] / DFEAT;
  const int E = in_sizes[1] / 2;

  float* ws = (float*)d_ws;
  float* emin = ws + OFF_EMIN; float* esum = ws + OFF_ESUM;
  float* emax = ws + OFF_EMAX; float* ecnt = ws + OFF_ECNT;
  float* vmin = ws + OFF_VMIN; float* vsum = ws + OFF_VSUM;
  float* vmax = ws + OFF_VMAX; float* vcnt = ws + OFF_VCNT;
  float* X = ws + OFF_X;
  float* H = ws + OFF_H;

  // 1) init accumulators (193*256 == 49408 exactly)
  init_stats_kernel<<<193, 256, 0, stream>>>(ws);

  // 2) edge stats: seg = batch[edgeij[0][e]]
  seg_reduce_kernel<<<256, 256, 0, stream>>>(e_attr, batch, edgeij, E,
                                             emin, esum, emax, ecnt);
  // 3) node stats: seg = batch[n]
  seg_reduce_kernel<<<128, 256, 0, stream>>>(v_attr, batch, nullptr, N,
                                             vmin, vsum, vmax, vcnt);

  // 4) assemble X [128, 576]
  build_x_kernel<<<32, 256, 0, stream>>>(gattr, ws, X);

  // 5) H = relu(X @ W1 + b1) : 128 waves (16 blocks x 8 waves)
  gemm1_relu_kernel<<<16, 256, 0, stream>>>(X, W1, b1, H);

  // 6) out = H @ W2 + b2 : 32 waves (4 blocks x 8 waves)
  gemm2_kernel<<<4, 256, 0, stream>>>(H, W2, b2, out);
}